// MEGUpdate_17549236371659
// MI455X (gfx1250) — compile-verified
//
#include <hip/hip_runtime.h>
#include <math.h>

// MEG update: out = expm( log(R) - ETA*(G - G^T) ), batched B=8192, N=64.
//
// Design for MI455X (compile-only reasoning):
//  - HBM floor: 3 x 8192 x 16KB = 400 MB -> ~17us @ 23.3 TB/s. All expm work
//    stays in LDS (single pass over HBM).
//  - f32 matrix core path: V_WMMA_F32_16X16X4_F32 keeps full fp32 precision
//    (series + 8 squarings amplify rounding; f16 would be risky, and we have
//    huge matrix-op headroom since total FLOPs ~60 GFLOP).
//  - LDS bandwidth is the on-chip limiter (K=4 shape => low intensity), so:
//      * 2x2 register blocking: each wave owns a 32x32 block (4 accum tiles);
//        per K-step 4 fragment loads feed 4 WMMAs (256B LDS traffic per WMMA,
//        half of the unblocked version). 4 waves = one 64x64 matmul.
//      * Fixed scaling 2^-8 + exactly 8 squarings (valid: ||U||inf <= 199 < 256
//        given R>=0.1 and eta*skew small) -> no norm pass, uniform control flow.
//      * Paterson-Stockmeyer for Taylor-13: A2 = A*A, then 6 Horner steps
//        T = T*A2 + (c_{2q} I + c_{2q+1} A), linear part fused into the
//        matmul epilogue. 1 + 6 + 8 = 15 matmuls total (vs 19 plain Taylor).
//  - 128 threads/WG (4 waves), 3 LDS matrices (52 KB) -> 6 WGs per 320KB WGP.

typedef __attribute__((ext_vector_type(2))) float v2f;
typedef __attribute__((ext_vector_type(8))) float v8f;

#define NMAT  64
#define STR   68          // 68 % 64 == 4 -> conflict-free banks, 8B-aligned pairs
#define ETA   0.1f
#define SCALE (1.0f / 256.0f)
#define NSQ   8           // fixed squaring count (2^-8 pre-scale)

// 1/k! for k = 0..13
__constant__ float INV_FACT[14] = {
    1.0f, 1.0f, 5.0e-1f, 1.6666666666666666e-1f, 4.1666666666666664e-2f,
    8.3333333333333332e-3f, 1.3888888888888889e-3f, 1.9841269841269841e-4f,
    2.4801587301587302e-5f, 2.7557319223985893e-6f, 2.7557319223985888e-7f,
    2.5052108385441720e-8f, 2.0876756987868098e-9f, 1.6059043836821613e-10f};

__device__ __forceinline__ v8f wmma4(v2f a, v2f b, v8f c) {
  // 8 args: (neg_a, A, neg_b, B, c_mod, C, reuse_a, reuse_b)
  return __builtin_amdgcn_wmma_f32_16x16x4_f32(false, a, false, b, (short)0, c,
                                               false, false);
}

// 32x32 block (2x2 tiles) of C = A*B over full K=64; operands in LDS.
// A-frag (16x4 f32): lanes 0-15 K={0,1}, lanes 16-31 K={2,3}, M = lane&15.
// B-frag (4x16):     N across lanes; lanes 0-15 rows K={0,1}, 16-31 K={2,3}.
__device__ __forceinline__ void mm64_2x2(const float* __restrict__ A,
                                         const float* __restrict__ B,
                                         int bi, int bj, int m, int khalf,
                                         v8f c[2][2]) {
#pragma unroll
  for (int kc = 0; kc < NMAT; kc += 4) {
    v2f a0, a1, b0, b1;
    const int ar = (bi * 32 + m) * STR + kc + khalf;
    a0.x = A[ar];                a0.y = A[ar + 1];
    a1.x = A[ar + 16 * STR];     a1.y = A[ar + 16 * STR + 1];
    const int br = (kc + khalf) * STR + bj * 32 + m;
    b0.x = B[br];                b0.y = B[br + STR];
    b1.x = B[br + 16];           b1.y = B[br + STR + 16];
    c[0][0] = wmma4(a0, b0, c[0][0]);
    c[0][1] = wmma4(a0, b1, c[0][1]);
    c[1][0] = wmma4(a1, b0, c[1][0]);
    c[1][1] = wmma4(a1, b1, c[1][1]);
  }
}

__device__ __forceinline__ void zero_acc(v8f c[2][2]) {
#pragma unroll
  for (int ii = 0; ii < 2; ++ii)
#pragma unroll
    for (int jj = 0; jj < 2; ++jj) c[ii][jj] = (v8f){0.f,0.f,0.f,0.f,0.f,0.f,0.f,0.f};
}

// C/D tile layout: VGPR v, lanes 0-15: M=v, N=lane; lanes 16-31: M=v+8, N=lane-16.
__device__ __forceinline__ void store_block(float* __restrict__ dst,
                                            v8f c[2][2], int bi, int bj,
                                            int lane) {
  const int vrow = (lane >> 4) << 3;
  const int m    = lane & 15;
#pragma unroll
  for (int ii = 0; ii < 2; ++ii)
#pragma unroll
    for (int jj = 0; jj < 2; ++jj) {
      const int r0 = bi * 32 + ii * 16 + vrow;
      const int cc = bj * 32 + jj * 16 + m;
#pragma unroll
      for (int v = 0; v < 8; ++v) dst[(r0 + v) * STR + cc] = c[ii][jj][v];
    }
}

__global__ __launch_bounds__(128) void megexp_kernel(
    const float* __restrict__ R, const float* __restrict__ G,
    float* __restrict__ out) {
  __shared__ float Alds[NMAT * STR];    // A = U * 2^-8
  __shared__ float A2lds[NMAT * STR];   // A^2
  __shared__ float Tlds[NMAT * STR];    // Horner accumulator / result

  const size_t base = (size_t)blockIdx.x * NMAT * NMAT;
  const int t     = threadIdx.x;
  const int lane  = t & 31;
  const int wave  = t >> 5;         // 0..3
  const int bi    = wave >> 1;      // 32-row block
  const int bj    = wave & 1;       // 32-col block
  const int m     = lane & 15;
  const int khalf = (lane >> 4) << 1;  // 0 or 2
  const int vrow  = (lane >> 4) << 3;  // 0 or 8

  // ---- A = (log(R) - ETA*(G - G^T)) * 2^-8 : single pass over HBM ----
  for (int e = t; e < NMAT * NMAT; e += 128) {
    const int i = e >> 6, j = e & 63;
    float u = logf(R[base + e]) -
              ETA * (G[base + e] - G[base + (size_t)j * NMAT + i]);
    Alds[i * STR + j] = u * SCALE;
  }
  __syncthreads();

  // ---- A2 = A*A ; concurrently T = c12*I + c13*A (disjoint buffers) ----
  {
    v8f c[2][2];
    zero_acc(c);
    mm64_2x2(Alds, Alds, bi, bj, m, khalf, c);
    store_block(A2lds, c, bi, bj, lane);
  }
  for (int e = t; e < NMAT * NMAT; e += 128) {
    const int i = e >> 6, j = e & 63;
    Tlds[i * STR + j] =
        INV_FACT[13] * Alds[i * STR + j] + ((i == j) ? INV_FACT[12] : 0.f);
  }
  __syncthreads();

  // ---- Paterson-Stockmeyer Horner: q = 5..0:  T = T*A2 + c_{2q} I + c_{2q+1} A
#pragma unroll
  for (int q = 5; q >= 0; --q) {
    v8f c[2][2];
    zero_acc(c);
    mm64_2x2(Tlds, A2lds, bi, bj, m, khalf, c);
    __syncthreads();                       // all reads of T complete
    const float c0 = INV_FACT[2 * q];
    const float c1 = INV_FACT[2 * q + 1];
#pragma unroll
    for (int ii = 0; ii < 2; ++ii)
#pragma unroll
      for (int jj = 0; jj < 2; ++jj) {
        const int r0 = bi * 32 + ii * 16 + vrow;
        const int cc = bj * 32 + jj * 16 + m;
#pragma unroll
        for (int v = 0; v < 8; ++v) {
          const int r = r0 + v;
          Tlds[r * STR + cc] = c[ii][jj][v] + c1 * Alds[r * STR + cc] +
                               ((r == cc) ? c0 : 0.f);
        }
      }
    __syncthreads();
  }

  // ---- 8 squarings: T = T*T  (uniform trip count across all blocks) ----
  for (int s = 0; s < NSQ; ++s) {
    v8f c[2][2];
    zero_acc(c);
    mm64_2x2(Tlds, Tlds, bi, bj, m, khalf, c);
    __syncthreads();                       // reads complete before write-back
    store_block(Tlds, c, bi, bj, lane);
    __syncthreads();
  }

  // ---- write result ----
  for (int e = t; e < NMAT * NMAT; e += 128) {
    out[base + e] = Tlds[(e >> 6) * STR + (e & 63)];
  }
}

extern "C" void kernel_launch(void* const* d_in, const int* in_sizes, int n_in,
                              void* d_out, int out_size, void* d_ws,
                              size_t ws_size, hipStream_t stream) {
  (void)n_in; (void)out_size; (void)d_ws; (void)ws_size;
  const float* R = (const float*)d_in[0];
  const float* G = (const float*)d_in[1];
  float* out = (float*)d_out;
  const int nbatch = in_sizes[0] / (NMAT * NMAT);  // 8192
  megexp_kernel<<<nbatch, 128, 0, stream>>>(R, G, out);
}